// TripletLoss_89129161326976
// MI455X (gfx1250) — compile-verified
//
#include <hip/hip_runtime.h>

#define BATCH 512
#define DIM 128
#define MARGIN 1.0f

typedef float v2f __attribute__((ext_vector_type(2)));
typedef float v8f __attribute__((ext_vector_type(8)));

// ---------------------------------------------------------------------------
// Kernel 1: per-row squared norms  sq[i] = sum_d E[i][d]^2
// ---------------------------------------------------------------------------
__global__ void sqnorm_kernel(const float* __restrict__ E, float* __restrict__ sq) {
    int row = blockIdx.x * blockDim.x + threadIdx.x;
    if (row < BATCH) {
        const float* e = E + row * DIM;
        float s = 0.0f;
#pragma unroll 8
        for (int d = 0; d < DIM; ++d) s += e[d] * e[d];
        sq[row] = s;
    }
}

// ---------------------------------------------------------------------------
// Kernel 2: Gram matrix via V_WMMA_F32_16X16X4_F32, fused distance epilogue.
// One wave (32 threads) computes one 16x16 tile of dist.
//   A tile = E[i0:i0+16, k:k+4]         (16x4,  M x K)
//   B tile = E^T[k:k+4, j0:j0+16]       (4x16,  K x N)  == rows of E at j0
// 32-bit A layout: lane L holds A[M=L%16][K=2*(L/16)+v], v=0,1 (2 VGPRs)
// B is the mirror with N=L%16 -> identical float2 load pattern from E.
// C/D layout: lane L, reg r holds C[M = r + 8*(L/16)][N = L%16]
// ---------------------------------------------------------------------------
__global__ void dist_wmma_kernel(const float* __restrict__ E,
                                 const float* __restrict__ sq,
                                 float* __restrict__ dist) {
    const int tile = blockIdx.x;         // 0 .. 1023
    const int ti = tile >> 5;            // row-tile
    const int tj = tile & 31;            // col-tile
    const int lane = threadIdx.x;        // 0 .. 31
    const int mr = lane & 15;
    const int koff = (lane >> 4) << 1;   // 0 or 2

    const float* arow = E + (ti * 16 + mr) * DIM + koff;
    const float* brow = E + (tj * 16 + mr) * DIM + koff;

    v8f c = {};
#pragma unroll
    for (int k = 0; k < DIM; k += 4) {
        v2f a = *(const v2f*)(arow + k);
        v2f b = *(const v2f*)(brow + k);
        // 8 args: (neg_a, A, neg_b, B, c_mod, C, reuse_a, reuse_b)
        c = __builtin_amdgcn_wmma_f32_16x16x4_f32(
                false, a, false, b, (short)0, c, false, false);
    }

    const int n = tj * 16 + mr;          // global column
    const float sn = sq[n];
    const int mbase = ti * 16 + ((lane >> 4) << 3);
#pragma unroll
    for (int r = 0; r < 8; ++r) {
        const int mi = mbase + r;        // global row
        float d2 = sq[mi] + sn - 2.0f * c[r];
        d2 = fmaxf(d2, 0.0f);            // clamp (also covers the >0 mask)
        if (mi == n) d2 = 0.0f;          // zero diagonal
        dist[mi * BATCH + n] = d2;
    }
}

// ---------------------------------------------------------------------------
// Kernel 3: semi-hard negative mining + per-anchor partial loss.
// One block per anchor i; distance row + negative mask cached in LDS.
// Only positive pairs (label match, j != i) do the O(B) candidate scan.
// ---------------------------------------------------------------------------
__global__ void semihard_kernel(const float* __restrict__ dist,
                                const int* __restrict__ labels,
                                float* __restrict__ psum,
                                float* __restrict__ pcnt) {
    __shared__ float sd[BATCH];
    __shared__ unsigned char sneg[BATCH];
    __shared__ float red[256];

    const int i = blockIdx.x;
    const int tid = threadIdx.x;
    const int li = labels[i];

    for (int c = tid; c < BATCH; c += 256) {
        sd[c] = dist[i * BATCH + c];
        sneg[c] = (labels[c] != li) ? 1 : 0;
    }
    __syncthreads();

    // row max / row min over the full row
    float lmax = -3.402823466e+38f, lmin = 3.402823466e+38f;
    for (int c = tid; c < BATCH; c += 256) {
        lmax = fmaxf(lmax, sd[c]);
        lmin = fminf(lmin, sd[c]);
    }
    red[tid] = lmax;
    __syncthreads();
    for (int s = 128; s > 0; s >>= 1) {
        if (tid < s) red[tid] = fmaxf(red[tid], red[tid + s]);
        __syncthreads();
    }
    const float rowMax = red[0];
    __syncthreads();
    red[tid] = lmin;
    __syncthreads();
    for (int s = 128; s > 0; s >>= 1) {
        if (tid < s) red[tid] = fminf(red[tid], red[tid + s]);
        __syncthreads();
    }
    const float rowMin = red[0];
    __syncthreads();

    // neg_inside = max_c (d[c]-rowMin)*negmask + rowMin  (0 included, as in ref)
    float lni = 0.0f;
    for (int c = tid; c < BATCH; c += 256) {
        float v = sneg[c] ? (sd[c] - rowMin) : 0.0f;
        lni = fmaxf(lni, v);
    }
    red[tid] = lni;
    __syncthreads();
    for (int s = 128; s > 0; s >>= 1) {
        if (tid < s) red[tid] = fmaxf(red[tid], red[tid + s]);
        __syncthreads();
    }
    const float negInside = red[0] + rowMin;
    __syncthreads();

    float localSum = 0.0f, localCnt = 0.0f;
    for (int j = tid; j < BATCH; j += 256) {
        const bool pos = (!sneg[j]) && (j != i);
        if (pos) {
            const float dj = sd[j];
            // neg_outside = min_c ((d[c]-rowMax)*cond) + rowMax, 0 included
            float m = 0.0f;
            bool exists = false;
            for (int c = 0; c < BATCH; ++c) {
                if (sneg[c] && (sd[c] > dj)) {
                    exists = true;
                    m = fminf(m, sd[c] - rowMax);
                }
            }
            const float neg = exists ? (m + rowMax) : negInside;
            const float term = MARGIN + dj - neg;
            localSum += fmaxf(term, 0.0f);
            localCnt += 1.0f;
        }
    }

    red[tid] = localSum;
    __syncthreads();
    for (int s = 128; s > 0; s >>= 1) {
        if (tid < s) red[tid] += red[tid + s];
        __syncthreads();
    }
    if (tid == 0) psum[i] = red[0];
    __syncthreads();
    red[tid] = localCnt;
    __syncthreads();
    for (int s = 128; s > 0; s >>= 1) {
        if (tid < s) red[tid] += red[tid + s];
        __syncthreads();
    }
    if (tid == 0) pcnt[i] = red[0];
}

// ---------------------------------------------------------------------------
// Kernel 4: deterministic final reduction (fixed order, no atomics).
// ---------------------------------------------------------------------------
__global__ void finalize_kernel(const float* __restrict__ psum,
                                const float* __restrict__ pcnt,
                                float* __restrict__ out) {
    if (blockIdx.x == 0 && threadIdx.x == 0) {
        float s = 0.0f, c = 0.0f;
        for (int i = 0; i < BATCH; ++i) { s += psum[i]; c += pcnt[i]; }
        out[0] = s / c;
    }
}

extern "C" void kernel_launch(void* const* d_in, const int* in_sizes, int n_in,
                              void* d_out, int out_size, void* d_ws, size_t ws_size,
                              hipStream_t stream) {
    const float* E = (const float*)d_in[0];       // [512,128] fp32
    const int* labels = (const int*)d_in[1];      // [512] int32

    float* ws = (float*)d_ws;
    float* dist = ws;                              // 512*512
    float* sq   = ws + BATCH * BATCH;              // 512
    float* psum = sq + BATCH;                      // 512
    float* pcnt = psum + BATCH;                    // 512

    sqnorm_kernel<<<(BATCH + 127) / 128, 128, 0, stream>>>(E, sq);
    dist_wmma_kernel<<<(BATCH / 16) * (BATCH / 16), 32, 0, stream>>>(E, sq, dist);
    semihard_kernel<<<BATCH, 256, 0, stream>>>(dist, labels, psum, pcnt);
    finalize_kernel<<<1, 32, 0, stream>>>(psum, pcnt, (float*)d_out);
}